// Attention_20340965114503
// MI455X (gfx1250) — compile-verified
//
#include <hip/hip_runtime.h>
#include <hip/hip_bf16.h>
#include <math.h>

// Shapes from the reference
#define BB 32
#define LL 2048
#define NN 1024

typedef __attribute__((ext_vector_type(2))) float v2f;
typedef __attribute__((ext_vector_type(8))) float v8f;

// ---------------------------------------------------------------------------
// Kernel 1: s = s_t_hat @ W_dec^T + b_dec   via V_WMMA_F32_16X16X4_F32
// One wave per 16x16 output tile. 2 M-tiles x 64 N-tiles = 128 waves.
// A layout (16x4 f32, ISA 7.12.2): lanes 0-15 -> M=lane, VGPR0/1 = K=k0,k0+1;
//                                  lanes 16-31 -> M=lane-16, VGPR0/1 = K=k0+2,k0+3.
// B (4x16) mirrors A with N in place of M. B[k,j] = W_dec[j*N + k] (row-major,
// contiguous in k) so both A and B are contiguous float2 loads.
// C/D layout: VGPR v: lanes0-15 row m0+v, col n0+lane; lanes16-31 row m0+v+8.
// ---------------------------------------------------------------------------
__global__ void dec_proj_wmma(const float* __restrict__ X,    // s_t_hat [B,N]
                              const float* __restrict__ W,    // W_dec   [N,N]
                              const float* __restrict__ bias, // b_dec   [N]
                              float* __restrict__ S) {        // out s   [B,N]
  const int lane  = threadIdx.x & 31;
  const int wave  = threadIdx.x >> 5;
  const int tile  = blockIdx.x * (blockDim.x >> 5) + wave;   // 0..127
  const int m0    = (tile >> 6) << 4;                        // 0 or 16
  const int n0    = (tile & 63) << 4;                        // 0..1008
  const int half  = lane >> 4;                               // 0 or 1
  const int idx16 = lane & 15;
  const int koff  = half << 1;                               // 0 or 2

  const float* __restrict__ Arow = X + (size_t)(m0 + idx16) * NN;  // A: row m over k
  const float* __restrict__ Brow = W + (size_t)(n0 + idx16) * NN;  // B(k,j)=W[j,k]: row j over k

  v8f c = {};
#pragma unroll 4
  for (int k0 = 0; k0 < NN; k0 += 4) {
    const float2 av = *(const float2*)(Arow + k0 + koff);
    const float2 bv = *(const float2*)(Brow + k0 + koff);
    v2f a, b;
    a.x = av.x; a.y = av.y;
    b.x = bv.x; b.y = bv.y;
    c = __builtin_amdgcn_wmma_f32_16x16x4_f32(
        /*neg_a=*/false, a, /*neg_b=*/false, b,
        /*c_mod=*/(short)0, c, /*reuse_a=*/false, /*reuse_b=*/false);
  }

  const int col = n0 + idx16;
  const float bcol = bias[col];
#pragma unroll
  for (int v = 0; v < 8; ++v) {
    const int row = m0 + v + (half << 3);
    S[(size_t)row * NN + col] = c[v] + bcol;
  }
}

// ---------------------------------------------------------------------------
// Kernel 2: e[b,l] = sum_n tanh(enc[b,l,n] + s[b,n] + cov[b,l]*Wc[n]) * Wv[n]
// One wave32 per (b,l) row; 8 x float4 per lane covers N=1024. Streams
// enc_feat (256MB) exactly once. s/Wc/Wv are hot in L2.
// ---------------------------------------------------------------------------
__global__ void energy_kernel(const float* __restrict__ enc,   // [B,L,N]
                              const float* __restrict__ S,     // [B,N]
                              const float* __restrict__ cover, // [B,L]
                              const float* __restrict__ Wc,    // [N]
                              const float* __restrict__ Wv,    // [N]
                              float* __restrict__ E) {         // [B,L]
  const int lane = threadIdx.x & 31;
  const int wave = (blockIdx.x * blockDim.x + threadIdx.x) >> 5; // 0..B*L-1
  const int b = wave >> 11;          // / 2048
  const int l = wave & (LL - 1);

  const float cov = cover[b * LL + l];
  const float* __restrict__ er = enc + ((size_t)b * LL + l) * NN;
  const float* __restrict__ sr = S + (size_t)b * NN;

  float acc = 0.f;
#pragma unroll
  for (int i = 0; i < 8; ++i) {
    const int n = (i * 32 + lane) * 4;
    const float4 e4 = *(const float4*)(er + n);
    const float4 s4 = *(const float4*)(sr + n);
    const float4 c4 = *(const float4*)(Wc + n);
    const float4 v4 = *(const float4*)(Wv + n);
    acc += tanhf(e4.x + s4.x + cov * c4.x) * v4.x;
    acc += tanhf(e4.y + s4.y + cov * c4.y) * v4.y;
    acc += tanhf(e4.z + s4.z + cov * c4.z) * v4.z;
    acc += tanhf(e4.w + s4.w + cov * c4.w) * v4.w;
  }
  // wave32 butterfly reduction
#pragma unroll
  for (int off = 16; off > 0; off >>= 1)
    acc += __shfl_xor(acc, off, 32);
  if (lane == 0) E[b * LL + l] = acc;
}

// ---------------------------------------------------------------------------
// Kernel 3: masked softmax + renorm over L, plus coverage_new.
// One 256-thread block per batch row (8 elems/thread).
// ---------------------------------------------------------------------------
__device__ __forceinline__ float blockMax(float v, volatile float* red) {
#pragma unroll
  for (int o = 16; o > 0; o >>= 1) v = fmaxf(v, __shfl_xor(v, o, 32));
  if ((threadIdx.x & 31) == 0) red[threadIdx.x >> 5] = v;
  __syncthreads();
  float r = red[0];
#pragma unroll
  for (int i = 1; i < 8; ++i) r = fmaxf(r, red[i]);
  __syncthreads();
  return r;
}

__device__ __forceinline__ float blockSum(float v, volatile float* red) {
#pragma unroll
  for (int o = 16; o > 0; o >>= 1) v += __shfl_xor(v, o, 32);
  if ((threadIdx.x & 31) == 0) red[threadIdx.x >> 5] = v;
  __syncthreads();
  float r = red[0];
#pragma unroll
  for (int i = 1; i < 8; ++i) r += red[i];
  __syncthreads();
  return r;
}

__global__ void softmax_kernel(const float* __restrict__ E,     // [B,L]
                               const float* __restrict__ mask,  // [B,L]
                               const float* __restrict__ cover, // [B,L]
                               float* __restrict__ attn,        // [B,L] (d_out)
                               float* __restrict__ covnew) {    // [B,L] (d_out)
  __shared__ float red[8];
  const int b = blockIdx.x;
  const int t = threadIdx.x;
  const float* __restrict__ e = E + b * LL;

  float vals[8];
  float m = -INFINITY;
#pragma unroll
  for (int i = 0; i < 8; ++i) {
    vals[i] = e[i * 256 + t];
    m = fmaxf(m, vals[i]);
  }
  m = blockMax(m, red);

  float ex[8];
  float Z = 0.f;
#pragma unroll
  for (int i = 0; i < 8; ++i) {
    ex[i] = __expf(vals[i] - m);
    Z += ex[i];
  }
  Z = blockSum(Z, red);

  float mk[8];
  float Sm = 0.f;
#pragma unroll
  for (int i = 0; i < 8; ++i) {
    mk[i] = mask[b * LL + i * 256 + t];
    Sm += ex[i] * mk[i];
  }
  Sm = blockSum(Sm, red);

  // attn = (ex*mk/Z) / (Sm/Z + 1e-12)
  const float inv = 1.0f / (Sm + 1e-12f * Z);
#pragma unroll
  for (int i = 0; i < 8; ++i) {
    const int idx = b * LL + i * 256 + t;
    const float a = ex[i] * mk[i] * inv;
    attn[idx] = a;
    covnew[idx] = cover[idx] + a;
  }
}

// ---------------------------------------------------------------------------
// Kernel 4: partial context. Block = (b, chunk c of 256 L-values), covers all
// N via float4/thread. attn chunk staged in LDS. Streams h (256MB) once.
// part[b][c][n], c in 0..7.
// ---------------------------------------------------------------------------
__global__ void context_partial(const float* __restrict__ h,    // [B,L,N]
                                const float* __restrict__ attn, // [B,L]
                                float* __restrict__ part) {     // [B,8,N]
  __shared__ float sa[256];
  const int b = blockIdx.x >> 3;
  const int c = blockIdx.x & 7;
  const int l0 = c * 256;
  sa[threadIdx.x] = attn[b * LL + l0 + threadIdx.x];
  __syncthreads();

  const int n = threadIdx.x * 4;
  const float* __restrict__ hp = h + ((size_t)b * LL + l0) * NN + n;
  float4 acc = {0.f, 0.f, 0.f, 0.f};
#pragma unroll 4
  for (int j = 0; j < 256; ++j) {
    const float w = sa[j];
    const float4 hv = *(const float4*)(hp + (size_t)j * NN);
    acc.x += w * hv.x;
    acc.y += w * hv.y;
    acc.z += w * hv.z;
    acc.w += w * hv.w;
  }
  *(float4*)(part + (size_t)(b * 8 + c) * NN + n) = acc;
}

// ---------------------------------------------------------------------------
// Kernel 5: reduce 8 partials -> context [B,N]
// ---------------------------------------------------------------------------
__global__ void context_reduce(const float* __restrict__ part, // [B,8,N]
                               float* __restrict__ ctx) {      // [B,N] (d_out)
  const int i = blockIdx.x * blockDim.x + threadIdx.x;  // 0..B*N-1
  const int b = i >> 10;
  const int n = i & (NN - 1);
  float acc = 0.f;
#pragma unroll
  for (int c = 0; c < 8; ++c) acc += part[(size_t)(b * 8 + c) * NN + n];
  ctx[i] = acc;
}

// ---------------------------------------------------------------------------
// Launch
// d_in order: h(0), enc_feat(1), attn_mask(2), s_t_hat(3), coverage(4),
//             W_dec(5), b_dec(6), W_c(7), W_v(8)
// d_out:  attn_dist [B*L] | context [B*N] | coverage_new [B*L]
// ws (floats): s [B*N] | e [B*L] | part [B*8*N]
// ---------------------------------------------------------------------------
extern "C" void kernel_launch(void* const* d_in, const int* in_sizes, int n_in,
                              void* d_out, int out_size, void* d_ws, size_t ws_size,
                              hipStream_t stream) {
  const float* h        = (const float*)d_in[0];
  const float* enc_feat = (const float*)d_in[1];
  const float* mask     = (const float*)d_in[2];
  const float* s_t_hat  = (const float*)d_in[3];
  const float* coverage = (const float*)d_in[4];
  const float* W_dec    = (const float*)d_in[5];
  const float* b_dec    = (const float*)d_in[6];
  const float* W_c      = (const float*)d_in[7];
  const float* W_v      = (const float*)d_in[8];

  float* out       = (float*)d_out;
  float* attn_out  = out;                 // B*L
  float* ctx_out   = out + BB * LL;       // B*N
  float* cov_out   = out + BB * LL + BB * NN;

  float* ws   = (float*)d_ws;
  float* s    = ws;                        // B*N   = 32768
  float* e    = ws + BB * NN;              // B*L   = 65536
  float* part = ws + BB * NN + BB * LL;    // B*8*N = 262144

  // 1) dec_proj via WMMA: 128 waves -> 32 blocks x 128 threads (4 waves each)
  dec_proj_wmma<<<32, 128, 0, stream>>>(s_t_hat, W_dec, b_dec, s);

  // 2) energy: one wave per (b,l): 65536 waves -> 8192 blocks x 256
  energy_kernel<<<(BB * LL) / 8, 256, 0, stream>>>(enc_feat, s, coverage, W_c, W_v, e);

  // 3) masked softmax + coverage update: one block per batch row
  softmax_kernel<<<BB, 256, 0, stream>>>(e, mask, coverage, attn_out, cov_out);

  // 4) context partials: B*8 = 256 blocks x 256
  context_partial<<<BB * 8, 256, 0, stream>>>(h, attn_out, part);

  // 5) reduce partials -> context: B*N/256 = 128 blocks
  context_reduce<<<(BB * NN) / 256, 256, 0, stream>>>(part, ctx_out);
}